// GCNII_41729902247981
// MI455X (gfx1250) — compile-verified
//
#include <hip/hip_runtime.h>
#include <math.h>

typedef __attribute__((ext_vector_type(2))) float v2f;
typedef __attribute__((ext_vector_type(8))) float v8f;

// ---------------------------------------------------------------------------
// Degree / normalization kernels
// ---------------------------------------------------------------------------
__global__ void k_fill_one(float* __restrict__ deg, int n) {
    int i = blockIdx.x * blockDim.x + threadIdx.x;
    if (i < n) deg[i] = 1.0f;                    // self-loop contributes 1
}

__global__ void k_degree(const int* __restrict__ col, float* __restrict__ deg, int e) {
    int i = blockIdx.x * blockDim.x + threadIdx.x;
    if (i < e) unsafeAtomicAdd(&deg[col[i]], 1.0f);
}

__global__ void k_rsqrt(float* __restrict__ deg, int n) {
    int i = blockIdx.x * blockDim.x + threadIdx.x;
    if (i < n) {
        float d = deg[i];
        deg[i] = d > 0.0f ? rsqrtf(d) : 0.0f;
    }
}

// ---------------------------------------------------------------------------
// Dense GEMM via V_WMMA_F32_16X16X4_F32, weights staged in LDS.
// OUT[16r x 64] = X[16r x K] @ W^T + bias, W is [64, K] row-major (torch
// Linear layout) so B[k][n] = W[n*K + k] -> contiguous v2f ds_load_b64.
// One wave per 16-row tile; A-fragments held in registers for all 4 col tiles.
// ---------------------------------------------------------------------------
template <int K, bool RELU, bool STOREPRE>
__global__ void k_dense(const float* __restrict__ X, const float* __restrict__ W,
                        const float* __restrict__ bias,
                        float* __restrict__ outA, float* __restrict__ outPre,
                        int ntiles) {
    __shared__ float wlds[64 * K];
    constexpr int NV4 = (64 * K) / 4;
    for (int i = threadIdx.x; i < NV4; i += 256)
        ((float4*)wlds)[i] = ((const float4*)W)[i];
    __syncthreads();

    int wid  = (blockIdx.x * blockDim.x + threadIdx.x) >> 5;
    int lane = threadIdx.x & 31;
    if (wid >= ntiles) return;                    // uniform per wave
    int ln = lane & 15;                           // row-in-tile for A, col for B
    int kh = lane >> 4;                           // K-half selector

    // Preload all A fragments (K/4 steps, v2f each)
    const float* xrow = X + (size_t)(wid * 16 + ln) * K + 2 * kh;
    v2f afr[K / 4];
#pragma unroll
    for (int kk = 0; kk < K / 4; ++kk)
        afr[kk] = *(const v2f*)(xrow + 4 * kk);

#pragma unroll
    for (int j = 0; j < 4; ++j) {
        const float* wrow = wlds + (size_t)(16 * j + ln) * K + 2 * kh;
        v8f acc = {};
#pragma unroll
        for (int kk = 0; kk < K / 4; ++kk) {
            v2f b = *(const v2f*)(wrow + 4 * kk);   // ds_load_b64
            acc = __builtin_amdgcn_wmma_f32_16x16x4_f32(
                false, afr[kk], false, b, (short)0, acc, false, false);
        }
        int n = 16 * j + ln;
        float bv = bias[n];
#pragma unroll
        for (int r = 0; r < 8; ++r) {
            int m = r + 8 * kh;
            size_t gi = (size_t)(wid * 16 + m) * 64 + n;
            float v = acc[r] + bv;
            if constexpr (STOREPRE) outPre[gi] = v;
            outA[gi] = RELU ? fmaxf(v, 0.0f) : v;
        }
    }
}

// ---------------------------------------------------------------------------
// Self-loop init: agg[i,f] = dinv[i]^2 * h[i,f]   (also resets agg each layer)
// ---------------------------------------------------------------------------
__global__ void k_selfloop(const float* __restrict__ dinv, const float* __restrict__ h,
                           float* __restrict__ agg, long long total) {
    long long i = (long long)blockIdx.x * blockDim.x + threadIdx.x;
    if (i >= total) return;
    int node = (int)(i >> 6);
    float d = dinv[node];
    agg[i] = d * d * h[i];
}

// ---------------------------------------------------------------------------
// Edge scatter: agg[col] += dinv[row]*dinv[col] * h[row].
// 16 lanes per edge, 4 floats per lane, FP32 hardware atomics (L2-resident).
// ---------------------------------------------------------------------------
__global__ void k_scatter(const int* __restrict__ row, const int* __restrict__ col,
                          const float* __restrict__ dinv, const float* __restrict__ h,
                          float* __restrict__ agg, int e) {
    long long t = (long long)blockIdx.x * blockDim.x + threadIdx.x;
    int ei = (int)(t >> 4);
    if (ei >= e) return;
    int f = (int)(t & 15) * 4;
    int r = row[ei], c = col[ei];
    float nrm = dinv[r] * dinv[c];
    const float4 hv = *(const float4*)(h + (size_t)r * 64 + f);
    float* dst = agg + (size_t)c * 64 + f;
    unsafeAtomicAdd(dst + 0, nrm * hv.x);
    unsafeAtomicAdd(dst + 1, nrm * hv.y);
    unsafeAtomicAdd(dst + 2, nrm * hv.z);
    unsafeAtomicAdd(dst + 3, nrm * hv.w);
}

// ---------------------------------------------------------------------------
// Fused GCNII layer: hh = 0.9*agg + 0.1*x0;
//   h = relu((1-beta)*hh + beta*(hh @ Wl)),  Wl [64,64] row-major:
//   B[k][n] = Wl[k*64+n]  (staged in LDS; b.x/b.y are two ds_load_b32)
// ---------------------------------------------------------------------------
__global__ void k_layer(const float* __restrict__ agg, const float* __restrict__ x0,
                        const float* __restrict__ Wl, float beta,
                        float* __restrict__ h, int ntiles) {
    __shared__ float wlds[64 * 64];
    for (int i = threadIdx.x; i < 1024; i += 256)
        ((float4*)wlds)[i] = ((const float4*)Wl)[i];
    __syncthreads();

    int wid  = (blockIdx.x * blockDim.x + threadIdx.x) >> 5;
    int lane = threadIdx.x & 31;
    if (wid >= ntiles) return;
    int ln = lane & 15;
    int kh = lane >> 4;

    size_t rb = (size_t)(wid * 16 + ln) * 64 + 2 * kh;
    v2f afr[16];
#pragma unroll
    for (int kk = 0; kk < 16; ++kk) {
        v2f ag = *(const v2f*)(agg + rb + 4 * kk);
        v2f xx = *(const v2f*)(x0 + rb + 4 * kk);
        afr[kk] = ag * 0.9f + xx * 0.1f;
    }
    float ob = 1.0f - beta;
#pragma unroll
    for (int j = 0; j < 4; ++j) {
        const float* wcol = wlds + 2 * kh * 64 + 16 * j + ln;
        v8f acc = {};
#pragma unroll
        for (int kk = 0; kk < 16; ++kk) {
            v2f b;
            b.x = wcol[(size_t)(4 * kk) * 64];        // ds_load_b32
            b.y = wcol[(size_t)(4 * kk + 1) * 64];    // ds_load_b32
            acc = __builtin_amdgcn_wmma_f32_16x16x4_f32(
                false, afr[kk], false, b, (short)0, acc, false, false);
        }
#pragma unroll
        for (int r = 0; r < 8; ++r) {
            int m = r + 8 * kh;
            int n = 16 * j + ln;
            size_t gi = (size_t)(wid * 16 + m) * 64 + n;
            float hh = 0.9f * agg[gi] + 0.1f * x0[gi];
            float v = ob * hh + beta * acc[r];
            h[gi] = fmaxf(v, 0.0f);
        }
    }
}

// ---------------------------------------------------------------------------
// Host launcher
// ---------------------------------------------------------------------------
extern "C" void kernel_launch(void* const* d_in, const int* in_sizes, int n_in,
                              void* d_out, int out_size, void* d_ws, size_t ws_size,
                              hipStream_t stream) {
    const float* x      = (const float*)d_in[0];
    const int*   ei     = (const int*)  d_in[1];
    const float* lin0_w = (const float*)d_in[2];
    const float* lin0_b = (const float*)d_in[3];
    const float* conv_w = (const float*)d_in[4];
    const float* lin1_w = (const float*)d_in[5];
    const float* lin1_b = (const float*)d_in[6];
    float* out = (float*)d_out;

    const int N = in_sizes[0] / 128;      // 100000
    const int E = in_sizes[1] / 2;        // 1600000
    const int* row = ei;                  // edge_index[0] = source
    const int* col = ei + E;              // edge_index[1] = target

    float* ws   = (float*)d_ws;
    float* dinv = ws;                               // N
    float* x0   = dinv + N;                         // N*64
    float* h    = x0 + (size_t)N * 64;              // N*64
    float* agg  = h  + (size_t)N * 64;              // N*64

    const int ntiles = N / 16;                      // N divisible by 16
    const int gemmGrid = (ntiles * 32 + 255) / 256;
    const long long NF = (long long)N * 64;

    // normalization
    k_fill_one<<<(N + 255) / 256, 256, 0, stream>>>(dinv, N);
    k_degree  <<<(E + 255) / 256, 256, 0, stream>>>(col, dinv, E);
    k_rsqrt   <<<(N + 255) / 256, 256, 0, stream>>>(dinv, N);

    // x0 = x @ lin0_w^T + b ; h = relu(x0)
    k_dense<128, true, true><<<gemmGrid, 256, 0, stream>>>(x, lin0_w, lin0_b, h, x0, ntiles);

    // 4 GCNII layers
    for (int l = 0; l < 4; ++l) {
        float beta = logf(0.5f / (float)(l + 1) + 1.0f);
        k_selfloop<<<(int)((NF + 255) / 256), 256, 0, stream>>>(dinv, h, agg, NF);
        k_scatter <<<(int)(((long long)E * 16 + 255) / 256), 256, 0, stream>>>(
            row, col, dinv, h, agg, E);
        k_layer   <<<gemmGrid, 256, 0, stream>>>(agg, x0, conv_w + (size_t)l * 64 * 64,
                                                 beta, h, ntiles);
    }

    // out = h @ lin1_w^T + b
    k_dense<64, false, false><<<gemmGrid, 256, 0, stream>>>(h, lin1_w, lin1_b, out, nullptr, ntiles);
}